// SparseLinear_56341380989458
// MI455X (gfx1250) — compile-verified
//
#include <hip/hip_runtime.h>

// ---------------------------------------------------------------------------
// SparseLinear (CSR W, 1% density) as densify + fp32 WMMA GEMM on gfx1250.
//   Y[M=8192, N=4096] = X[8192, 4096] @ Wdense^T,  Wdense built in d_ws.
// Memory-bound (~270 MB mandatory traffic @ 23.3 TB/s ~= 12 us); dense W
// (64 MB) stays L2-resident (192 MB); f32 WMMA compute floor ~1 us.
// GLOBAL_LOAD_ASYNC_TO_LDS_B128 double-buffered staging (ASYNCcnt pipeline)
// replaces the serialized load->wait->ds_store path; Y uses NT stores.
// ---------------------------------------------------------------------------

typedef __attribute__((ext_vector_type(2))) float v2f;
typedef __attribute__((ext_vector_type(4))) float vf4;
typedef __attribute__((ext_vector_type(8))) float v8f;
typedef __attribute__((ext_vector_type(4))) int   v4i;

#define IN_DIM   4096
#define OUT_DIM  4096
#define KSTEP    16
#define LDS_STR  20          // 16 + 4 pad: 16B-aligned rows, conflict-free frags
#define NCHUNK   (IN_DIM / KSTEP)

#if defined(__gfx1250__) && __has_builtin(__builtin_amdgcn_global_load_async_to_lds_b128)
#define HAVE_ASYNC_LDS 1
#else
#define HAVE_ASYNC_LDS 0
#endif

static __device__ __forceinline__ void async_cp16(const float* g, float* l) {
#if HAVE_ASYNC_LDS
    // Builtin takes non-const generic pointers to 16B vectors (per hipcc
    // diagnostic: param type is 'int __vector(4) *').
    __builtin_amdgcn_global_load_async_to_lds_b128(
        (v4i*)g, (v4i*)l, /*offset=*/0, /*cpol=*/0);
#else
    *(vf4*)l = *(const vf4*)g;           // sync fallback
#endif
}

static __device__ __forceinline__ void wait_async_le_8() {
#if HAVE_ASYNC_LDS
#if __has_builtin(__builtin_amdgcn_s_wait_asynccnt)
    __builtin_amdgcn_s_wait_asynccnt(8);
#else
    asm volatile("s_wait_asynccnt 0x8" ::: "memory");
#endif
#endif
}

static __device__ __forceinline__ void wait_async_le_0() {
#if HAVE_ASYNC_LDS
#if __has_builtin(__builtin_amdgcn_s_wait_asynccnt)
    __builtin_amdgcn_s_wait_asynccnt(0);
#else
    asm volatile("s_wait_asynccnt 0x0" ::: "memory");
#endif
#endif
}

// ---- Phase 1a: zero the dense-W scratch (64 MB) ---------------------------
__global__ __launch_bounds__(256) void zero_w_kernel(vf4* __restrict__ w4, int n4) {
    int stride = gridDim.x * blockDim.x;
    for (int i = blockIdx.x * blockDim.x + threadIdx.x; i < n4; i += stride) {
        vf4 z = {0.f, 0.f, 0.f, 0.f};
        w4[i] = z;
    }
}

// ---- Phase 1b: scatter CSR nonzeros into dense W (row-major OUT x IN) -----
__global__ __launch_bounds__(64) void scatter_kernel(float* __restrict__ Wd,
                                                     const float* __restrict__ values,
                                                     const int* __restrict__ row_offs,
                                                     const int* __restrict__ col_idx) {
    int row = blockIdx.x;
    int beg = row_offs[row];
    int end = row_offs[row + 1];
    for (int j = beg + (int)threadIdx.x; j < end; j += 64) {
        Wd[(size_t)row * IN_DIM + col_idx[j]] = values[j];
    }
}

// ---- Phase 2: dense GEMM with V_WMMA_F32_16X16X4_F32 ----------------------
// Block: 128 threads = 4 waves. Block tile 128(M) x 128(N).
// Wave (wm,wn) in 2x2 owns a 64x64 tile = 4x4 grid of 16x16 WMMA tiles.
// K staged in 16-wide chunks, double-buffered via async-to-LDS DMA.

// Stage one 128x16 tile of X and W into LDS: 8 b128 async copies per thread.
static __device__ __forceinline__ void prefetch_chunk(const float* __restrict__ X,
                                                      const float* __restrict__ Wd,
                                                      float* xs, float* ws,
                                                      int m0, int n0, int kb, int tid) {
#pragma unroll
    for (int it = 0; it < 4; ++it) {
        int idx = it * 128 + tid;
        int row = idx >> 2;          // 0..127
        int seg = idx & 3;           // 16B segment within the 16-col row
        async_cp16(X  + (size_t)(m0 + row) * IN_DIM + kb + seg * 4,
                   xs + row * LDS_STR + seg * 4);
        async_cp16(Wd + (size_t)(n0 + row) * IN_DIM + kb + seg * 4,
                   ws + row * LDS_STR + seg * 4);
    }
}

__global__ __launch_bounds__(128)
void spmm_wmma_kernel(const float* __restrict__ X,
                      const float* __restrict__ Wd,
                      float* __restrict__ Y) {
    __shared__ float Xs[2][128 * LDS_STR];   // 2 x 10 KB
    __shared__ float Ws[2][128 * LDS_STR];   // 2 x 10 KB  (40 KB total)

    const int tid  = threadIdx.x;
    const int lane = tid & 31;
    const int wave = tid >> 5;
    const int wm   = wave >> 1;          // 0..1 : M half of block tile
    const int wn   = wave & 1;           // 0..1 : N half of block tile
    const int m0   = blockIdx.y * 128;
    const int n0   = blockIdx.x * 128;

    const int lm = lane & 15;            // M/N index within 16-wide tile
    const int kh = (lane >> 4) * 2;      // lanes 0-15 -> K0/K1, 16-31 -> K2/K3

    v8f acc[4][4];
#pragma unroll
    for (int t = 0; t < 4; ++t)
#pragma unroll
        for (int u = 0; u < 4; ++u) {
            v8f z = {0.f, 0.f, 0.f, 0.f, 0.f, 0.f, 0.f, 0.f};
            acc[t][u] = z;
        }

    // Pipeline prologue: DMA chunk 0 into buffer 0.
    prefetch_chunk(X, Wd, &Xs[0][0], &Ws[0][0], m0, n0, 0, tid);

    for (int ic = 0; ic < NCHUNK; ++ic) {
        const int cur = ic & 1;

        // Kick off the next chunk's DMA into the other buffer (safe: that
        // buffer's consumers all passed the previous trailing barrier).
        if (ic + 1 < NCHUNK) {
            prefetch_chunk(X, Wd, &Xs[cur ^ 1][0], &Ws[cur ^ 1][0],
                           m0, n0, (ic + 1) * KSTEP, tid);
            wait_async_le_8();           // chunk ic's 8 copies done; 8 newer in flight
        } else {
            wait_async_le_0();
        }
        __syncthreads();                 // all waves' DMA for chunk ic visible

        const float* xs = &Xs[cur][0];
        const float* ws = &Ws[cur][0];
#pragma unroll
        for (int k4 = 0; k4 < KSTEP; k4 += 4) {
            v2f a[4], b[4];
#pragma unroll
            for (int t = 0; t < 4; ++t) {
                int m = wm * 64 + t * 16 + lm;   // A: 16x4, lanes = M rows
                a[t] = *(const v2f*)(&xs[m * LDS_STR + k4 + kh]);
                int n = wn * 64 + t * 16 + lm;   // B: 4x16, lanes = N cols
                b[t] = *(const v2f*)(&ws[n * LDS_STR + k4 + kh]);
            }
#pragma unroll
            for (int t = 0; t < 4; ++t)
#pragma unroll
                for (int u = 0; u < 4; ++u)
                    acc[t][u] = __builtin_amdgcn_wmma_f32_16x16x4_f32(
                        /*neg_a=*/false, a[t],
                        /*neg_b=*/false, b[u],
                        /*c_mod=*/(short)0, acc[t][u],
                        /*reuse_a=*/false, /*reuse_b=*/false);
        }
        __syncthreads();                 // compute done before buffer reuse
    }

    // Store C (non-temporal: Y is write-once, keep W/X resident in L2).
    // VGPR r holds M = r (lanes 0-15) / r+8 (lanes 16-31); lane%16 = N.
#pragma unroll
    for (int t = 0; t < 4; ++t) {
#pragma unroll
        for (int u = 0; u < 4; ++u) {
            int n     = n0 + wn * 64 + u * 16 + lm;
            int mbase = m0 + wm * 64 + t * 16 + (lane >> 4) * 8;
#pragma unroll
            for (int r = 0; r < 8; ++r) {
                __builtin_nontemporal_store(acc[t][u][r],
                                            &Y[(size_t)(mbase + r) * OUT_DIM + n]);
            }
        }
    }
}

// ---------------------------------------------------------------------------
extern "C" void kernel_launch(void* const* d_in, const int* in_sizes, int n_in,
                              void* d_out, int out_size, void* d_ws, size_t ws_size,
                              hipStream_t stream) {
    const float* x        = (const float*)d_in[0];  // (BATCH, SEQ, IN) fp32
    const float* values   = (const float*)d_in[1];  // (nnz,) fp32
    const int*   row_offs = (const int*)d_in[2];    // (OUT+1,) int32
    const int*   col_idx  = (const int*)d_in[3];    // (nnz,) int32

    float* Wd = (float*)d_ws;                       // dense (OUT, IN) fp32, 64 MB
    float* y  = (float*)d_out;                      // (BATCH, SEQ, OUT) fp32

    const int M = in_sizes[0] / IN_DIM;             // 8192 = BATCH*SEQ

    // 1) Build dense W in workspace (stream-ordered ahead of the GEMM).
    const int n4 = (OUT_DIM * IN_DIM) / 4;
    zero_w_kernel<<<4096, 256, 0, stream>>>((vf4*)Wd, n4);
    scatter_kernel<<<OUT_DIM, 64, 0, stream>>>(Wd, values, row_offs, col_idx);

    // 2) Dense fp32 WMMA GEMM: Y = X @ Wd^T.
    dim3 grid(OUT_DIM / 128, M / 128);              // (32, 64)
    spmm_wmma_kernel<<<grid, 128, 0, stream>>>(x, Wd, y);
}